// FastMaskedConv2D_82678120448547
// MI455X (gfx1250) — compile-verified
//
#include <hip/hip_runtime.h>

typedef __attribute__((ext_vector_type(2))) float v2f;
typedef __attribute__((ext_vector_type(8))) float v8f;
typedef int vi4 __attribute__((vector_size(16)));   // 16-byte payload for async LDS copy

#define AS1 __attribute__((address_space(1)))
#define AS3 __attribute__((address_space(3)))

#define BATCH   4096
#define IN_F    64
#define OUT_F   64
#define LW      64
#define KH      4
#define KW      7
#define NTAPS   24   // autoregressive mask (EXCLUSIVE): kh<3 all 7 kw, kh==3 only kw<3

#if __has_builtin(__builtin_amdgcn_global_load_async_to_lds_b128)
#define HAVE_ASYNC_LDS 1
#endif

// ---------------------------------------------------------------------------
// Kernel 1: cache pass-through + autoregressive update (memory-bound, float4).
// XLA dynamic_update_slice clamps the (-1) row start to row 0:
//   index<=0          : copy
//   index%64 != 0     : copy, but [b, 0, (index-1)%64, :] = inputs[b, :]
//   index%64 == 0     : row shift (row r = old row r+1, last row = 0)
// ---------------------------------------------------------------------------
__global__ __launch_bounds__(256) void cache_update_kernel(
    const float* __restrict__ inputs,
    const float* __restrict__ cache,
    const int*   __restrict__ pindex,
    float*       __restrict__ cache_out)
{
  unsigned idx = blockIdx.x * 256u + threadIdx.x;     // float4 index, 16,777,216 total
  const int index = *pindex;
  const unsigned ci4 = idx & 15u;                     // float4 within IN_F row
  const unsigned w   = (idx >> 4) & 63u;              // width column
  const unsigned r   = (idx >> 10) & 3u;              // receptive-height row
  const unsigned b   = idx >> 12;                     // batch
  const float4* __restrict__ src = (const float4*)cache;
  float4 v;
  if (index < 1) {
    v = src[idx];
  } else if ((index & 63) != 0) {
    if (r == 0u && w == (unsigned)((index - 1) & 63)) {
      v = ((const float4*)inputs)[b * 16u + ci4];
    } else {
      v = src[idx];
    }
  } else {
    if (r < 3u) v = src[idx + 1024u];                 // 64*64/4 float4 per (b, row)
    else        v = make_float4(0.f, 0.f, 0.f, 0.f);
  }
  ((float4*)cache_out)[idx] = v;
}

// ---------------------------------------------------------------------------
// Kernel 2: masked-conv as GEMM with V_WMMA_F32_16X16X4_F32.
// Block = 128 threads (4 waves); block tile = 16 batch rows x 64 out features;
// each wave owns one 16x16 tile. A window staged into LDS via
// GLOBAL_LOAD_ASYNC_TO_LDS_B128 (zero-padded columns via plain ds stores),
// B from global; K = 24 taps * 64 ci contracted 4 at a time with two
// interleaved accumulators to break the WMMA D->C dependency chain.
// ---------------------------------------------------------------------------
__global__ __launch_bounds__(128) void conv_wmma_kernel(
    const float* __restrict__ weights,   // (KH, KW, IN_F, OUT_F)
    const float* __restrict__ bias,      // (OUT_F)
    const int*   __restrict__ pindex,
    const float* __restrict__ cacheU,    // updated cache (BATCH, KH, LW, IN_F)
    float*       __restrict__ y)         // (BATCH, OUT_F)
{
  __shared__ float As[NTAPS * 16 * IN_F];   // 24 taps * 16 rows * 64 ci = 98304 B

  const int tid   = threadIdx.x;
  const int lane  = tid & 31;
  const int wave  = tid >> 5;              // n-tile
  const int mbase = blockIdx.x << 4;       // 16 batch rows per block

  const int index   = *pindex;
  const int index_w = ((index % LW) + LW) % LW;   // Python-style modulo

  // ---- stage A window into LDS; zero for width columns outside [0, LW) ----
  float4* As4 = (float4*)As;
  for (int j = 0; j < 48; ++j) {
    int fidx = j * 128 + tid;              // 0..6143 float4 slots
    int ci4  = fidx & 15;
    int m    = (fidx >> 4) & 15;
    int s    = fidx >> 8;                  // tap slot 0..23
    int kh   = (s < 21) ? (s / 7) : 3;
    int kw   = (s < 21) ? (s % 7) : (s - 21);
    int wcol = index_w + kw - (KW / 2);
    if (wcol >= 0 && wcol < LW) {
      size_t g = ((size_t)(mbase + m) * KH + kh) * LW + (size_t)wcol;
#ifdef HAVE_ASYNC_LDS
      __builtin_amdgcn_global_load_async_to_lds_b128(
          (AS1 vi4*)((const float4*)cacheU + g * 16 + ci4),
          (AS3 vi4*)(As4 + fidx), 0, 0);
#else
      As4[fidx] = ((const float4*)cacheU)[g * 16 + ci4];
#endif
    } else {
      As4[fidx] = make_float4(0.f, 0.f, 0.f, 0.f);
    }
  }
#ifdef HAVE_ASYNC_LDS
#if __has_builtin(__builtin_amdgcn_s_wait_asynccnt)
  __builtin_amdgcn_s_wait_asynccnt(0);
#else
  asm volatile("s_wait_asynccnt 0x0" ::: "memory");
#endif
#endif
  __syncthreads();

  // ---- two accumulators; c0 carries the bias ----
  const int nloc = lane & 15;              // N within tile; also A's M row
  const int n    = (wave << 4) + nloc;
  const int koff = (lane >> 4) << 1;       // K pair: 0 (lanes 0-15) / 2 (16-31)
  v8f c0, c1;
  {
    float bv = bias[n];
    #pragma unroll
    for (int r = 0; r < 8; ++r) { c0[r] = bv; c1[r] = 0.f; }
  }

  // ---- contract K = 24 taps * 64 ci, 4 per WMMA (f32 in, f32 out) ----
  for (int t = 0; t < NTAPS; ++t) {
    int kh = (t < 21) ? (t / 7) : 3;
    int kw = (t < 21) ? (t % 7) : (t - 21);
    const float* ap = &As[(t * 16 + nloc) * IN_F];
    const float* bp = weights + (size_t)((kh * KW + kw) * IN_F) * OUT_F + n;
    #pragma unroll
    for (int ci = 0; ci < IN_F; ci += 8) {
      v2f a0 = *(const v2f*)(ap + ci + koff);            // ds_load_b64
      v2f a1 = *(const v2f*)(ap + ci + 4 + koff);
      v2f b0, b1;
      b0.x = bp[(size_t)(ci + koff)     * OUT_F];
      b0.y = bp[(size_t)(ci + koff + 1) * OUT_F];
      b1.x = bp[(size_t)(ci + 4 + koff)     * OUT_F];
      b1.y = bp[(size_t)(ci + 4 + koff + 1) * OUT_F];
      c0 = __builtin_amdgcn_wmma_f32_16x16x4_f32(false, a0, false, b0,
                                                 (short)0, c0, false, false);
      c1 = __builtin_amdgcn_wmma_f32_16x16x4_f32(false, a1, false, b1,
                                                 (short)0, c1, false, false);
    }
  }
  v8f c = c0 + c1;

  // ---- store 16x16 tile: VGPR r holds M = r + 8*(lane>=16), N = n ----
  const int mrow0 = mbase + ((lane >> 4) << 3);
  #pragma unroll
  for (int r = 0; r < 8; ++r)
    y[(size_t)(mrow0 + r) * OUT_F + n] = c[r];
}

// ---------------------------------------------------------------------------
extern "C" void kernel_launch(void* const* d_in, const int* in_sizes, int n_in,
                              void* d_out, int out_size, void* d_ws, size_t ws_size,
                              hipStream_t stream) {
  (void)in_sizes; (void)n_in; (void)out_size; (void)d_ws; (void)ws_size;
  const float* inputs  = (const float*)d_in[0];
  const float* cache   = (const float*)d_in[1];
  const float* weights = (const float*)d_in[2];
  const float* bias    = (const float*)d_in[3];
  const int*   pindex  = (const int*)d_in[4];

  float* y        = (float*)d_out;                           // (4096, 64)
  float* cacheOut = (float*)d_out + (size_t)BATCH * OUT_F;   // (4096, 4, 64, 64)

  const unsigned nF4 = (unsigned)BATCH * KH * LW * IN_F / 4; // 16,777,216 float4
  cache_update_kernel<<<nF4 / 256, 256, 0, stream>>>(inputs, cache, pindex, cacheOut);
  conv_wmma_kernel<<<BATCH / 16, 128, 0, stream>>>(weights, bias, pindex, cacheOut, y);
}